// VectorQuantizer_37314675867753
// MI455X (gfx1250) — compile-verified
//
#include <hip/hip_runtime.h>
#include <hip/hip_bf16.h>
#include <math.h>

typedef __attribute__((ext_vector_type(2))) float v2f;
typedef __attribute__((ext_vector_type(8))) float v8f;
typedef __attribute__((__vector_size__(16))) int i32x4;   // matches builtin param type

#define B_SZ 32768
#define D_SZ 512
#define K_SZ 8192
#define BM   128          // rows per workgroup (resident in LDS)
#define BN   64           // codes per K-tile
#define DC   64           // d-chunk of codebook staged per buffer
#define DP   516          // padded LDS row stride (dwords) for A tile -> conflict-free
#define BNP  68           // padded LDS row stride (dwords) for B tile -> conflict-free
#define NDC  (D_SZ / DC)  // 8 chunks per K-tile
#define NCHUNK ((K_SZ / BN) * NDC)  // 1024 chunks total

#define GLOBAL_AS __attribute__((address_space(1)))
#define LDS_AS    __attribute__((address_space(3)))

#if __has_builtin(__builtin_amdgcn_global_load_async_to_lds_b128) && \
    __has_builtin(__builtin_amdgcn_s_wait_asynccnt)
#define HAVE_ASYNC 1
#else
#define HAVE_ASYNC 0
#endif

__device__ __forceinline__ GLOBAL_AS i32x4* gl_cast16(const void* p) {
  return (GLOBAL_AS i32x4*)(unsigned long long)p;             // p1 is 64-bit
}
__device__ __forceinline__ LDS_AS i32x4* lds_cast16(void* p) {
  return (LDS_AS i32x4*)(unsigned int)(unsigned long long)p;  // p3 is 32-bit (LDS offset)
}

// ---------------------------------------------------------------------------
// Kernel 1: e2[k] = sum(codebook[k]^2); zero counts + loss accumulator.
// ---------------------------------------------------------------------------
__global__ __launch_bounds__(256) void vq_e2_init(const float* __restrict__ cb,
                                                  float* __restrict__ e2,
                                                  unsigned* __restrict__ counts,
                                                  double* __restrict__ lossSum) {
  const int k = blockIdx.x * 256 + threadIdx.x;
  const float4* row = (const float4*)(cb + ((size_t)k << 9));
  float s = 0.f;
#pragma unroll 4
  for (int i = 0; i < D_SZ / 4; ++i) {
    float4 v = row[i];
    s += v.x * v.x + v.y * v.y + v.z * v.z + v.w * v.w;
  }
  e2[k] = s;
  counts[k] = 0u;
  if (k == 0) *lossSum = 0.0;
}

// ---------------------------------------------------------------------------
// Kernel 2: fused fp32-WMMA distance GEMM + running argmin per row.
//   S = a @ cb^T via V_WMMA_F32_16X16X4_F32
//   dist = (a2[m] + e2[n]) - 2*S   (exact reference op order)
// Codebook chunks are double-buffered through LDS with async-to-LDS loads
// (ASYNCcnt) so global latency hides behind the WMMA stream.
// ---------------------------------------------------------------------------
__global__ __launch_bounds__(256) void vq_argmin(const float* __restrict__ a,
                                                 const float* __restrict__ cb,
                                                 const float* __restrict__ e2,
                                                 float* __restrict__ outIdx,
                                                 unsigned* __restrict__ counts) {
  __shared__ __align__(16) float sA[BM * DP];       // 264192 B: 128x512 act tile
  __shared__ __align__(16) float sB[2 * BN * BNP];  //  34816 B: double-buffered chunk
  __shared__ float sA2[BM];                         // per-row |a|^2

  const int t    = threadIdx.x;
  const int wave = t >> 5;
  const int lane = t & 31;
  const int m    = lane & 15;      // fragment row/col index
  const int hh   = lane >> 4;      // half-wave selector (K pair / M half)
  const int m0   = blockIdx.x * BM;

  // ---- stage activation tile (coalesced b128 loads) ----
#pragma unroll 4
  for (int i = 0; i < 64; ++i) {
    int e = i * 1024 + t * 4;
    int r = e >> 9;
    int c = e & 511;
    float4 v = *(const float4*)(a + (size_t)(m0 + r) * D_SZ + c);
    *(float4*)(&sA[r * DP + c]) = v;
  }

  // chunk q covers codes [ (q/NDC)*BN , +BN ) and d-cols [ (q%NDC)*DC , +DC )
  auto issue_chunk = [&](int q) {
    const int nt  = q >> 3;         // q / NDC
    const int dcc = q & (NDC - 1);  // q % NDC
    const float* srcBase = cb + (size_t)(nt * BN) * D_SZ + dcc * DC;
    float* dstBase = &sB[(q & 1) * (BN * BNP)];
#pragma unroll
    for (int i = 0; i < 4; ++i) {
      int e = i * 1024 + t * 4;
      int r = e >> 6;   // 64 floats per chunk row
      int c = e & 63;
      const float* src = srcBase + (size_t)r * D_SZ + c;
      float* dst = dstBase + r * BNP + c;
#if HAVE_ASYNC
      __builtin_amdgcn_global_load_async_to_lds_b128(gl_cast16(src), lds_cast16(dst), 0, 0);
#else
      *(float4*)dst = *(const float4*)src;
      if (q + 1 < NCHUNK) __builtin_prefetch(src + DC, 0, 3);
#endif
    }
  };

  __syncthreads();

  // ---- per-row a2 for this wave's 16 rows ----
  {
    const float* rowp = &sA[(wave * 16 + m) * DP + hh * 256];
    float part = 0.f;
    for (int c = 0; c < 256; ++c) part += rowp[c] * rowp[c];
    part += __shfl_xor(part, 16);
    if (hh == 0) sA2[wave * 16 + m] = part;
  }
  __syncthreads();

  float a2r[8];
#pragma unroll
  for (int v = 0; v < 8; ++v) a2r[v] = sA2[wave * 16 + hh * 8 + v];

  float bestV[8];
  int   bestI[8];
#pragma unroll
  for (int v = 0; v < 8; ++v) { bestV[v] = 3.402823466e+38f; bestI[v] = 0; }

#if HAVE_ASYNC
  issue_chunk(0);  // prime the pipeline
#endif

  for (int nt = 0; nt < K_SZ / BN; ++nt) {
    const int n0 = nt * BN;
    v8f acc[4];
#pragma unroll
    for (int tt = 0; tt < 4; ++tt)
#pragma unroll
      for (int v = 0; v < 8; ++v) acc[tt][v] = 0.f;

    for (int dcc = 0; dcc < NDC; ++dcc) {
      const int q = nt * NDC + dcc;
#if HAVE_ASYNC
      __builtin_amdgcn_s_wait_asynccnt(0);     // own loads for chunk q landed
      __syncthreads();                         // everyone's landed; q-1 readers done
      if (q + 1 < NCHUNK) issue_chunk(q + 1);  // fill the other buffer
#else
      __syncthreads();
      issue_chunk(q);
      __syncthreads();
#endif
      const float* aRow  = &sA[(wave * 16 + m) * DP + dcc * DC + hh * 2];
      const float* bBase = &sB[(q & 1) * (BN * BNP)];
#pragma unroll
      for (int d4 = 0; d4 < DC / 4; ++d4) {
        const int dl = d4 * 4;
        v2f av;
        av.x = aRow[dl];
        av.y = aRow[dl + 1];
#pragma unroll
        for (int tt = 0; tt < 4; ++tt) {
          const float* bRow = bBase + (tt * 16 + m) * BNP + dl + hh * 2;
          v2f bv;
          bv.x = bRow[0];
          bv.y = bRow[1];
          acc[tt] = __builtin_amdgcn_wmma_f32_16x16x4_f32(
              false, av, false, bv, (short)0, acc[tt], false, false);
        }
      }
    }

    // ---- epilogue: dist + running argmin (strict < keeps first index) ----
#pragma unroll
    for (int tt = 0; tt < 4; ++tt) {
      const int n = n0 + tt * 16 + m;
      const float e2v = e2[n];
#pragma unroll
      for (int v = 0; v < 8; ++v) {
        float dist = (a2r[v] + e2v) - 2.0f * acc[tt][v];
        if (dist < bestV[v]) { bestV[v] = dist; bestI[v] = n; }
      }
    }
  }

  // ---- cross-lane reduction within each 16-lane group (tie -> lower index) ----
#pragma unroll
  for (int off = 8; off >= 1; off >>= 1) {
#pragma unroll
    for (int v = 0; v < 8; ++v) {
      float ov = __shfl_xor(bestV[v], off);
      int   oi = __shfl_xor(bestI[v], off);
      if (ov < bestV[v] || (ov == bestV[v] && oi < bestI[v])) {
        bestV[v] = ov;
        bestI[v] = oi;
      }
    }
  }
  if (m == 0) {  // lanes 0 and 16 each own 8 rows (M = v + 8*hh)
#pragma unroll
    for (int v = 0; v < 8; ++v) {
      int row = m0 + wave * 16 + hh * 8 + v;
      outIdx[row] = (float)bestI[v];
      atomicAdd(&counts[bestI[v]], 1u);
    }
  }
}

// ---------------------------------------------------------------------------
// Kernel 3: z = codebook[idx]; accumulate sum((a - z_q)^2) into double.
// ---------------------------------------------------------------------------
__global__ __launch_bounds__(256) void vq_gather_loss(const float* __restrict__ a,
                                                      const float* __restrict__ cb,
                                                      const float* __restrict__ idxF,
                                                      float* __restrict__ z,
                                                      double* __restrict__ lossSum) {
  __shared__ float red[8];
  const int t = threadIdx.x;
  float acc = 0.f;
  size_t base = (size_t)blockIdx.x * 256 + t;  // float4 index
#pragma unroll
  for (int i = 0; i < 4; ++i) {
    size_t e   = base + (size_t)i * (256u * 4096u);
    size_t row = e >> 7;  // 128 float4 per row
    size_t c4  = e & 127;
    int k = (int)idxF[row];
    float4 q  = *(const float4*)(cb + ((size_t)k << 9) + (c4 << 2));
    float4 av = *(const float4*)(a + (e << 2));
    *(float4*)(z + (e << 2)) = q;
    float dx = av.x - q.x, dy = av.y - q.y, dz = av.z - q.z, dw = av.w - q.w;
    acc += dx * dx + dy * dy + dz * dz + dw * dw;
  }
#pragma unroll
  for (int off = 16; off >= 1; off >>= 1) acc += __shfl_xor(acc, off);
  if ((t & 31) == 0) red[t >> 5] = acc;
  __syncthreads();
  if (t == 0) {
    float b = 0.f;
    for (int w = 0; w < 8; ++w) b += red[w];
    atomicAdd(lossSum, (double)b);
  }
}

// ---------------------------------------------------------------------------
// Kernel 4: usage stats + losses.
// ---------------------------------------------------------------------------
__global__ __launch_bounds__(256) void vq_stats(const unsigned* __restrict__ counts,
                                                const double* __restrict__ lossSum,
                                                float* __restrict__ outScalars) {
  __shared__ float r0[8], r1[8], r2[8];
  const int t = threadIdx.x;
  float ent = 0.f, used = 0.f, usum = 0.f;
  for (int i = 0; i < K_SZ / 256; ++i) {
    int k = t + i * 256;
    float usage = (float)counts[k] * (1.0f / (float)B_SZ);
    ent -= usage * logf(usage + 1e-10f);
    used += (counts[k] > 0u) ? 1.0f : 0.0f;
    usum += usage;
  }
#pragma unroll
  for (int off = 16; off >= 1; off >>= 1) {
    ent  += __shfl_xor(ent, off);
    used += __shfl_xor(used, off);
    usum += __shfl_xor(usum, off);
  }
  if ((t & 31) == 0) { r0[t >> 5] = ent; r1[t >> 5] = used; r2[t >> 5] = usum; }
  __syncthreads();
  if (t == 0) {
    float E = 0.f, U = 0.f, S = 0.f;
    for (int w = 0; w < 8; ++w) { E += r0[w]; U += r1[w]; S += r2[w]; }
    float mse = (float)(*lossSum / (double)((size_t)B_SZ * (size_t)D_SZ));
    outScalars[0] = 0.25f * mse;               // commit_loss (BETA = 0.25)
    outScalars[1] = mse;                       // codebook_loss
    outScalars[2] = expf(-E);                  // perplexity
    outScalars[3] = U;                         // used_codes
    outScalars[4] = S * (1.0f / (float)K_SZ);  // usage_mean
  }
}

// ---------------------------------------------------------------------------
extern "C" void kernel_launch(void* const* d_in, const int* in_sizes, int n_in,
                              void* d_out, int out_size, void* d_ws, size_t ws_size,
                              hipStream_t stream) {
  (void)in_sizes; (void)n_in; (void)out_size; (void)ws_size;
  const float* a  = (const float*)d_in[0];
  const float* cb = (const float*)d_in[1];

  float* out     = (float*)d_out;
  float* outIdx  = out;                                   // [B] (idx as float)
  float* z       = out + B_SZ;                            // [B*D]
  float* scalars = out + B_SZ + (size_t)B_SZ * D_SZ;      // [5]

  float*    e2      = (float*)d_ws;                                 // K floats
  unsigned* counts  = (unsigned*)((char*)d_ws + (size_t)K_SZ * 4);  // K uints
  double*   lossSum = (double*)((char*)d_ws + (size_t)2 * K_SZ * 4);

  vq_e2_init<<<K_SZ / 256, 256, 0, stream>>>(cb, e2, counts, lossSum);
  vq_argmin<<<B_SZ / BM, 256, 0, stream>>>(a, cb, e2, outIdx, counts);
  vq_gather_loss<<<4096, 256, 0, stream>>>(a, cb, outIdx, z, lossSum);
  vq_stats<<<1, 256, 0, stream>>>(counts, lossSum, scalars);
}